// sampleFFNN_AR_13125420057076
// MI455X (gfx1250) — compile-verified
//
#include <hip/hip_runtime.h>
#include <hip/hip_bf16.h>

// Problem constants (match reference)
#define BB   4096
#define TT   512
#define FF   63
#define NI   64     // F + 1 feedback column
#define HH   1024
#define ROWS 16     // batch rows per workgroup
#define NWAVE 16    // waves per workgroup (512 threads, wave32)
#define TILES 4     // N-tiles (16 cols each) per wave: 16 waves * 4 * 16 = 1024 = HH
#define KB   16     // K blocks of 4: 16*4 = 64 = NI
#define NTHREADS (32 * NWAVE)

typedef float v2f __attribute__((ext_vector_type(2)));
typedef float v8f __attribute__((ext_vector_type(8)));

__global__ __launch_bounds__(NTHREADS, 1)
void ffnn_ar_scan_kernel(const float* __restrict__ in,    // (B,T,F)
                         const float* __restrict__ W1,    // (64,1024)
                         const float* __restrict__ b1,    // (1024)
                         const float* __restrict__ W2,    // (1024,1)
                         const float* __restrict__ b2,    // (1)
                         float* __restrict__ out)         // (B,T,1)
{
    // Double-buffered A tile (16x64, stride 68 dwords -> conflict-free b64 frag loads)
    __shared__ float lds_mi[2][ROWS][68];
    __shared__ float lds_part[NWAVE][ROWS];   // per-wave partial dot results

    const int tid  = threadIdx.x;
    const int wv   = tid >> 5;
    const int lane = tid & 31;
    const int ln   = lane & 15;               // N index within tile / M row for A
    const int khalf = (lane < 16) ? 0 : 2;    // f32 A/B fragment K offset per lane half
    const int b0   = blockIdx.x * ROWS;       // batch row base

    // ---- Preload this wave's W1 fragments (held in VGPRs for the whole scan) ----
    v2f   bf[TILES][KB];
    float b1v[TILES];
    float w2v[TILES];
#pragma unroll
    for (int ti = 0; ti < TILES; ++ti) {
        const int n = (wv * TILES + ti) * 16 + ln;
        b1v[ti] = b1[n];
        w2v[ti] = W2[n];
#pragma unroll
        for (int kb = 0; kb < KB; ++kb) {
            const int k = kb * 4 + khalf;
            v2f b;
            b.x = W1[k * HH + n];
            b.y = W1[(k + 1) * HH + n];
            bf[ti][kb] = b;
        }
    }

    // ---- Per-thread staging addresses: elements e0 = tid, e1 = tid + 512 of the 16x63 tile ----
    const int e0 = tid;
    const int e1 = tid + NTHREADS;
    const int m0 = e0 / FF, f0 = e0 - m0 * FF;
    const int m1 = e1 / FF, f1 = e1 - m1 * FF;
    const bool has1 = (e1 < ROWS * FF);
    const float* p0 = in + (size_t)(b0 + m0) * TT * FF + f0;
    const float* p1 = in + (size_t)(b0 + m1) * TT * FF + f1;
    const float* rowp = (tid < ROWS) ? (in + (size_t)(b0 + tid) * TT * FF) : in;

    const float b2v = b2[0];

    // ---- Init: out[:,0] = 0; pred(t=0)=0 fed into buffer parity of t=1 ----
    if (tid < ROWS) {
        out[(size_t)(b0 + tid) * TT] = 0.0f;
        lds_mi[1][tid][FF] = 0.0f;            // t=1 uses buffer 1, col 63 = prev_pred
    }

    // ---- Preload x(t=1) into registers ----
    float xr0 = p0[(size_t)1 * FF];
    float xr1 = has1 ? p1[(size_t)1 * FF] : 0.0f;

    __syncthreads();

    for (int t = 1; t < TT; ++t) {
        const int pbuf = t & 1;

        // ---- Stage registers -> LDS tile (cols 0..62; col 63 already written by finalizer) ----
        lds_mi[pbuf][m0][f0] = xr0;
        if (has1) lds_mi[pbuf][m1][f1] = xr1;

        // ---- Issue next step's global loads now; wait lands at next iteration's store ----
        {
            const size_t toff = (size_t)((t + 1 < TT) ? (t + 1) : t) * FF;
            xr0 = p0[toff];
            xr1 = has1 ? p1[toff] : 0.0f;
        }
        if (tid < ROWS && t + 2 < TT)         // keep t+2 rows L2-hot (global_prefetch_b8)
            __builtin_prefetch(rowp + (size_t)(t + 2) * FF, 0, 0);

        __syncthreads();                       // barrier A: mi[pbuf] complete

        // ---- z = mi @ W1 + b1 via V_WMMA_F32_16X16X4_F32 ----
        v8f acc[TILES];
#pragma unroll
        for (int ti = 0; ti < TILES; ++ti) {
#pragma unroll
            for (int r = 0; r < 8; ++r) acc[ti][r] = b1v[ti];
        }
#pragma unroll
        for (int kb = 0; kb < KB; ++kb) {
            v2f a;
            a.x = lds_mi[pbuf][ln][kb * 4 + khalf];
            a.y = lds_mi[pbuf][ln][kb * 4 + khalf + 1];
#pragma unroll
            for (int ti = 0; ti < TILES; ++ti) {
                acc[ti] = __builtin_amdgcn_wmma_f32_16x16x4_f32(
                    /*neg_a=*/false, a, /*neg_b=*/false, bf[ti][kb],
                    /*c_mod=*/(short)0, acc[ti], /*reuse_a=*/false, /*reuse_b=*/false);
            }
        }

        // ---- Fused ReLU + partial dot with W2 (each lane: 8 rows, 1 col per tile) ----
        float psum[8];
#pragma unroll
        for (int r = 0; r < 8; ++r) psum[r] = 0.0f;
#pragma unroll
        for (int ti = 0; ti < TILES; ++ti) {
#pragma unroll
            for (int r = 0; r < 8; ++r)
                psum[r] += fmaxf(acc[ti][r], 0.0f) * w2v[ti];
        }
        // Butterfly reduce across the 16 N-lanes of each half-wave
#pragma unroll
        for (int mask = 1; mask < 16; mask <<= 1) {
#pragma unroll
            for (int r = 0; r < 8; ++r)
                psum[r] += __shfl_xor(psum[r], mask, 32);
        }
        if (ln == 0) {
            const int rbase = (lane < 16) ? 0 : 8;
#pragma unroll
            for (int r = 0; r < 8; ++r)
                lds_part[wv][rbase + r] = psum[r];
        }
        __syncthreads();                       // barrier B: lds_part complete

        // ---- Cross-wave reduce -> pred; store + feed into NEXT parity buffer's col 63 ----
        if (tid < ROWS) {
            float s = b2v;
#pragma unroll
            for (int w = 0; w < NWAVE; ++w) s += lds_part[w][tid];
            out[(size_t)(b0 + tid) * TT + t] = s;
            lds_mi[pbuf ^ 1][tid][FF] = s;     // disjoint from cols 0..62 staged next iter
        }
        // no third barrier: next iteration's barrier A orders these writes vs WMMA reads
    }
}

extern "C" void kernel_launch(void* const* d_in, const int* in_sizes, int n_in,
                              void* d_out, int out_size, void* d_ws, size_t ws_size,
                              hipStream_t stream) {
    const float* in = (const float*)d_in[0];
    // d_in[1] = labels (unused by the forward reference)
    const float* W1 = (const float*)d_in[2];
    const float* b1 = (const float*)d_in[3];
    const float* W2 = (const float*)d_in[4];
    const float* b2 = (const float*)d_in[5];
    float* out = (float*)d_out;

    dim3 grid(BB / ROWS);     // 256 workgroups, one 16-row batch tile each
    dim3 block(NTHREADS);     // 512 threads = 16 wave32 waves
    ffnn_ar_scan_kernel<<<grid, block, 0, stream>>>(in, W1, b1, W2, b2, out);
}